// GraphGenerator_1322849927810
// MI455X (gfx1250) — compile-verified
//
#include <hip/hip_runtime.h>

// float2 / float8 ext-vector types for the f32 WMMA operands
typedef __attribute__((ext_vector_type(2))) float v2f;
typedef __attribute__((ext_vector_type(8))) float v8f;

#define RADIUS2 (0.1f * 0.1f)
#define COL_TILES_PER_WAVE 4   // each wave computes a 16x64 strip of the adjacency

// D = A x B with A row i = (x, y, z, |p_i|^2), B col j = (-2x, -2y, -2z, 1)
// => D[i][j] = |p_i|^2 - 2 p_i.p_j ;  adj <=> D[i][j] <= r^2 - |p_j|^2
__global__ __launch_bounds__(256) void radius_graph_wmma(
    const float* __restrict__ pos, const float* __restrict__ dom,
    float* __restrict__ out, int N, int tilesPerRow) {
  const int lane = threadIdx.x & 31;
  const int wave = threadIdx.x >> 5;
  const int colGroups = tilesPerRow / COL_TILES_PER_WAVE;  // 128
  const long long strip = (long long)blockIdx.x * 8 + wave;
  const int tileRow = (int)(strip / colGroups);
  const int cg      = (int)(strip % colGroups);
  if (tileRow >= tilesPerRow) return;  // whole wave exits together (EXEC uniform)

  const int idx15 = lane & 15;  // row index (A) / col index (B,C,D) within tile
  const int half  = lane >> 4;  // selects K-pair {0,1} or {2,3}

  const float dx = dom[0], dy = dom[1], dz = dom[2];
  const float factors[4] = {0.0f, 0.25f, 0.5f, 0.75f};

  // ---- Row-tile side: load once, build per-pass A operands in registers ----
  const int rowPt = tileRow * 16 + idx15;
  const float rx = pos[rowPt * 3 + 0];
  const float ry = pos[rowPt * 3 + 1];
  const float rz = pos[rowPt * 3 + 2];

  v2f aOp[4];
#pragma unroll
  for (int p = 0; p < 4; ++p) {
    const float f = factors[p];
    float srx = rx + f * dx; srx = (srx >= dx) ? srx - dx : srx;
    float sry = ry + f * dy; sry = (sry >= dy) ? sry - dy : sry;
    float srz = rz + f * dz; srz = (srz >= dz) ? srz - dz : srz;
    const float sqr = srx * srx + sry * sry + srz * srz;  // |p_row|^2
    // A operand layout (32-bit A 16x4): half 0 -> (K0,K1)=(x,y); half 1 -> (K2,K3)=(z,|p|^2)
    aOp[p].x = half ? srz : srx;
    aOp[p].y = half ? sqr : sry;
  }

  // ---- Sweep 4 column tiles, reusing the row-side operands ----
#pragma unroll
  for (int ct = 0; ct < COL_TILES_PER_WAVE; ++ct) {
    const int tileCol = cg * COL_TILES_PER_WAVE + ct;
    const int colPt = tileCol * 16 + idx15;
    const float cx = pos[colPt * 3 + 0];
    const float cy = pos[colPt * 3 + 1];
    const float cz = pos[colPt * 3 + 2];

    v8f acc;
#pragma unroll
    for (int v = 0; v < 8; ++v) acc[v] = 0.0f;

#pragma unroll
    for (int p = 0; p < 4; ++p) {
      const float f = factors[p];
      float scx = cx + f * dx; scx = (scx >= dx) ? scx - dx : scx;
      float scy = cy + f * dy; scy = (scy >= dy) ? scy - dy : scy;
      float scz = cz + f * dz; scz = (scz >= dz) ? scz - dz : scz;
      const float sqc = scx * scx + scy * scy + scz * scz;  // |p_col|^2

      // B operand: half 0 -> (-2x, -2y) ; half 1 -> (-2z, 1)
      v2f b;
      b.x = half ? (-2.0f * scz) : (-2.0f * scx);
      b.y = half ? 1.0f          : (-2.0f * scy);

      v8f d;
#pragma unroll
      for (int v = 0; v < 8; ++v) d[v] = 0.0f;

      d = __builtin_amdgcn_wmma_f32_16x16x4_f32(
          false, aOp[p], false, b, (short)0, d, false, false);

      const float thresh = RADIUS2 - sqc;  // per-column threshold (col = lane&15)
#pragma unroll
      for (int v = 0; v < 8; ++v) acc[v] = (d[v] <= thresh) ? 1.0f : acc[v];
    }

    // Clear diagonal (LOOP = False)
    const bool diagTile = (tileRow == tileCol);
#pragma unroll
    for (int v = 0; v < 8; ++v) {
      const int m = v + 8 * half;  // row within tile for this C/D VGPR
      acc[v] = (diagTile && (m == idx15)) ? 0.0f : acc[v];
    }

    // Non-temporal stores: 256 MB streaming output > 192 MB L2, don't thrash it
#pragma unroll
    for (int v = 0; v < 8; ++v) {
      const long long gr = (long long)(tileRow * 16 + v + 8 * half);
      float* p = out + gr * N + (long long)(tileCol * 16 + idx15);
      __builtin_nontemporal_store(acc[v], p);
    }
  }
}

extern "C" void kernel_launch(void* const* d_in, const int* in_sizes, int n_in,
                              void* d_out, int out_size, void* d_ws, size_t ws_size,
                              hipStream_t stream) {
  (void)n_in; (void)out_size; (void)d_ws; (void)ws_size;
  const float* pos = (const float*)d_in[0];
  const float* dom = (const float*)d_in[1];
  float* out = (float*)d_out;

  const int N = in_sizes[0] / 3;  // 8192
  const int tiles = N / 16;       // 512
  const int colGroups = tiles / COL_TILES_PER_WAVE;            // 128
  const long long strips = (long long)tiles * colGroups;       // 65536
  const int blocks = (int)((strips + 7) / 8);                  // 8 waves per block

  radius_graph_wmma<<<blocks, 256, 0, stream>>>(pos, dom, out, N, tiles);
}